// SpatialTransformer_41446434406563
// MI455X (gfx1250) — compile-verified
//
#include <hip/hip_runtime.h>

// Trilinear spatial transformer (dense displacement field), MI455X / gfx1250.
// B=2, D=160, H=192, W=160, C=1. Memory-bound gather kernel:
//  - loc_shift / out streamed once -> non-temporal 128-bit vmem ops
//  - volume (39 MB total) stays resident in the 192 MB L2 -> regular-hint gathers
//  - speculative global_prefetch_b8 on the displacement stream
//  - 3D launch mapping: zero integer divides in the index math

typedef float v4f __attribute__((ext_vector_type(4)));

constexpr int B_ = 2, D_ = 160, H_ = 192, W_ = 160;
constexpr int HW_   = H_ * W_;        // 30720
constexpr int DHW_  = D_ * HW_;       // 4,915,200
constexpr int YT_   = 8;              // y-rows per block
constexpr int XQ_   = W_ / 4;         // 40 x-quads per row
// block = 40 x 8 = 320 threads = 10 wave32 waves; each block covers a fully
// contiguous run of 1280 voxels (15360 B of the displacement stream).

__global__ __launch_bounds__(XQ_ * YT_)
void st_trilinear_kernel(const float* __restrict__ vol,
                         const float* __restrict__ shift,
                         float* __restrict__ out) {
  // ---- division-free coordinates -------------------------------------------
  const int x  = (int)threadIdx.x * 4;                 // 0..156
  const int y  = (int)blockIdx.x * YT_ + (int)threadIdx.y;  // 0..191
  const int zz = (int)blockIdx.y;                      // fused b*D + z, 0..319
  const int b  = (zz >= D_) ? 1 : 0;
  const int z  = zz - b * D_;

  const int v0 = (zz * H_ + y) * W_ + x;               // linear voxel index

  // ---- streamed displacement: 48 bytes, 16B-aligned, non-temporal ----------
  const float* sp = shift + (size_t)v0 * 3;
  const v4f sA = __builtin_nontemporal_load((const v4f*)(sp));
  const v4f sB = __builtin_nontemporal_load((const v4f*)(sp + 4));
  const v4f sC = __builtin_nontemporal_load((const v4f*)(sp + 8));
  const float s[12] = {sA[0], sA[1], sA[2], sA[3],
                       sB[0], sB[1], sB[2], sB[3],
                       sC[0], sC[1], sC[2], sC[3]};

  // speculative prefetch of the displacement stream ~3 blocks ahead
  // (gfx1250 global_prefetch_b8; locality 3 -> fill near caches too)
  __builtin_prefetch((const char*)sp + 49152, 0, 3);

  const float* __restrict__ vb = vol + (size_t)b * DHW_;

  v4f o;
#pragma unroll
  for (int j = 0; j < 4; ++j) {
    // sample coordinate = identity grid + displacement, clamped to the domain
    const float fz = fminf(fmaxf((float)z       + s[3 * j + 0], 0.0f), (float)(D_ - 1));
    const float fy = fminf(fmaxf((float)y       + s[3 * j + 1], 0.0f), (float)(H_ - 1));
    const float fx = fminf(fmaxf((float)(x + j) + s[3 * j + 2], 0.0f), (float)(W_ - 1));

    const int z0 = (int)fz, y0 = (int)fy, x0 = (int)fx;  // f>=0 -> trunc==floor
    const float dz = fz - (float)z0;
    const float dy = fy - (float)y0;
    const float dx = fx - (float)x0;
    const int z1 = min(z0 + 1, D_ - 1);
    const int y1 = min(y0 + 1, H_ - 1);
    const int x1 = min(x0 + 1, W_ - 1);

    // 4 row base pointers, 8 gathers (L2-resident volume, regular hint)
    const float* r00 = vb + (size_t)z0 * HW_ + (size_t)y0 * W_;
    const float* r01 = vb + (size_t)z0 * HW_ + (size_t)y1 * W_;
    const float* r10 = vb + (size_t)z1 * HW_ + (size_t)y0 * W_;
    const float* r11 = vb + (size_t)z1 * HW_ + (size_t)y1 * W_;

    const float v000 = r00[x0], v001 = r00[x1];
    const float v010 = r01[x0], v011 = r01[x1];
    const float v100 = r10[x0], v101 = r10[x1];
    const float v110 = r11[x0], v111 = r11[x1];

    // trilinear lerp tree (7 lerps, FMA form)
    const float c00 = fmaf(dx, v001 - v000, v000);
    const float c01 = fmaf(dx, v011 - v010, v010);
    const float c10 = fmaf(dx, v101 - v100, v100);
    const float c11 = fmaf(dx, v111 - v110, v110);
    const float c0  = fmaf(dy, c01 - c00, c00);
    const float c1  = fmaf(dy, c11 - c10, c10);
    o[j]            = fmaf(dz, c1 - c0, c0);
  }

  // ---- streamed output: 16 bytes, aligned, non-temporal --------------------
  __builtin_nontemporal_store(o, (v4f*)(out + v0));
}

extern "C" void kernel_launch(void* const* d_in, const int* in_sizes, int n_in,
                              void* d_out, int out_size, void* d_ws, size_t ws_size,
                              hipStream_t stream) {
  const float* vol   = (const float*)d_in[0];   // [B,D,H,W,1] f32
  const float* shift = (const float*)d_in[1];   // [B,D,H,W,3] f32
  float* outp        = (float*)d_out;           // [B,D,H,W,1] f32

  dim3 block(XQ_, YT_, 1);                      // 40 x 8 = 320 threads
  dim3 grid(H_ / YT_, B_ * D_, 1);              // 24 x 320 blocks
  st_trilinear_kernel<<<grid, block, 0, stream>>>(vol, shift, outp);
}